// PromptPool_58531814310368
// MI455X (gfx1250) — compile-verified
//
#include <hip/hip_runtime.h>
#include <math.h>

typedef __attribute__((ext_vector_type(2))) float v2f;
typedef __attribute__((ext_vector_type(4))) float f4;
typedef __attribute__((ext_vector_type(8))) float v8f;

#define EMBED 768
#define POOL  32
#define PLEN  32
#define NCTXL 32
#define NCLS  1000
#define SUFL  12
#define TOPK  5

// ---------------------------------------------------------------------------
// One wave computes dots[m] = dot(mat_row_m, vec) for m = 0..15 (len % 4 == 0)
// using V_WMMA_F32_16X16X4_F32 with the vector in column 0 of B.
//   A 16x4 f32 layout: lanes 0-15 hold K=0,1 ; lanes 16-31 hold K=2,3 (M = lane%16)
//   B 4x16 f32 layout: column N=0 lives in lanes 0 (K=0,1) and 16 (K=2,3)
//   D column 0: lane 0 -> M=0..7 in acc[0..7], lane 16 -> M=8..15
// ---------------------------------------------------------------------------
__device__ inline v8f wmma_matvec16(const float* __restrict__ mat, int ld,
                                    const float* __restrict__ vec, int len) {
  const int lane = threadIdx.x & 31;
  const int m    = lane & 15;
  const int koff = (lane >> 4) << 1;           // 0 or 2
  const float colsel = (m == 0) ? 1.0f : 0.0f; // only column 0 of B is non-zero
  v8f acc = {0.f,0.f,0.f,0.f,0.f,0.f,0.f,0.f};
  for (int kb = 0; kb < len; kb += 4) {
    v2f a  = *(const v2f*)(mat + m * ld + kb + koff);
    v2f bv = *(const v2f*)(vec + kb + koff);
    v2f b;
    b.x = bv.x * colsel;
    b.y = bv.y * colsel;
    // 8 args: (neg_a, A, neg_b, B, c_mod, C, reuse_a, reuse_b)
    acc = __builtin_amdgcn_wmma_f32_16x16x4_f32(false, a, false, b,
                                                (short)0, acc, false, false);
  }
  return acc;
}

// Scatter column 0 of D (16 dots) into dst16[0..15].
__device__ inline void wmma_col0_store(v8f acc, float* dst16) {
  const int lane = threadIdx.x & 31;
  if (lane == 0 || lane == 16) {
    const int mb = (lane == 16) ? 8 : 0;
#pragma unroll
    for (int r = 0; r < 8; ++r) dst16[mb + r] = acc[r];
  }
}

// ---------------------------------------------------------------------------
// Kernel 1: q = mean(x_embed, axis=0); q /= ||q||      (1 block, 256 threads)
// ---------------------------------------------------------------------------
__global__ void qnorm_kernel(const float* __restrict__ x, float* __restrict__ q) {
  __shared__ float qt[EMBED];
  __shared__ float red[8];
  __shared__ float invn;
  const int tid = threadIdx.x;
  float ss = 0.f;
  for (int d = tid; d < EMBED; d += 256) {
    float s = 0.f;
#pragma unroll
    for (int f = 0; f < 16; ++f) s += x[f * EMBED + d];
    s *= (1.0f / 16.0f);
    qt[d] = s;
    ss += s * s;
  }
#pragma unroll
  for (int o = 16; o > 0; o >>= 1) ss += __shfl_xor(ss, o, 32);
  if ((tid & 31) == 0) red[tid >> 5] = ss;
  __syncthreads();
  if (tid == 0) {
    float t = 0.f;
#pragma unroll
    for (int i = 0; i < 8; ++i) t += red[i];
    invn = 1.0f / sqrtf(t);
  }
  __syncthreads();
  for (int d = tid; d < EMBED; d += 256) q[d] = qt[d] * invn;
}

// ---------------------------------------------------------------------------
// Kernel 2: sim[p] = dot(key_p, q)/||key_p|| via WMMA, then top-5 select.
// 2 waves (64 threads); wave w handles keys 16w..16w+15.
// ---------------------------------------------------------------------------
__global__ void sim_topk_kernel(const float* __restrict__ key,
                                const float* __restrict__ q,
                                int* __restrict__ top_idx) {
  __shared__ float raw[POOL];
  __shared__ float sims[POOL];
  const int tid  = threadIdx.x;
  const int wave = tid >> 5;
  const int lane = tid & 31;
  const float* base = key + wave * 16 * EMBED;

  // per-row sum of squares (lane pair l / l+16 split even/odd elements)
  const int m   = lane & 15;
  const int off = lane >> 4;
  float ss = 0.f;
  for (int j = off; j < EMBED; j += 2) {
    float v = base[m * EMBED + j];
    ss += v * v;
  }
  ss += __shfl_xor(ss, 16, 32);

  // raw dots via WMMA
  v8f acc = wmma_matvec16(base, EMBED, q, EMBED);
  wmma_col0_store(acc, &raw[wave * 16]);
  __syncthreads();

  if (lane < 16) sims[wave * 16 + m] = raw[wave * 16 + m] * (1.0f / sqrtf(ss));
  __syncthreads();

  // top-5 selection (train_flag scaling is a positive constant: order-invariant)
  if (tid == 0) {
    unsigned chosen = 0u;
#pragma unroll
    for (int s = 0; s < TOPK; ++s) {
      int bi = 0; float bv = -3.0e38f;
      for (int p = 0; p < POOL; ++p)
        if (!((chosen >> p) & 1u) && sims[p] > bv) { bv = sims[p]; bi = p; }
      chosen |= (1u << bi);
      top_idx[s] = bi;
    }
  }
}

// ---------------------------------------------------------------------------
// Kernel 3: gates[k][l] = sigmoid(dot(pool[idx_k, l, :], alpha_w) + alpha_b)
// 10 waves (320 threads); wave c handles (k = c/2, tokens (c&1)*16 .. +15).
// ---------------------------------------------------------------------------
__global__ void gates_kernel(const float* __restrict__ pool,
                             const float* __restrict__ aw,
                             const float* __restrict__ ab,
                             const int* __restrict__ idx,
                             float* __restrict__ gates) {
  const int wave  = threadIdx.x >> 5;
  const int lane  = threadIdx.x & 31;
  const int k     = wave >> 1;
  const int lbase = (wave & 1) * 16;
  const float* base = pool + (size_t)idx[k] * (PLEN * EMBED) + lbase * EMBED;

  v8f acc = wmma_matvec16(base, EMBED, aw, EMBED);

  const float b = ab[0];
  if (lane == 0 || lane == 16) {
    const int mb = (lane == 16) ? 8 : 0;
#pragma unroll
    for (int r = 0; r < 8; ++r) {
      float d = acc[r] + b;
      gates[k * PLEN + lbase + mb + r] = 1.0f / (1.0f + __expf(-d));
    }
  }
}

// ---------------------------------------------------------------------------
// Kernel 4: combined[l, :] = sum_k gates[k][l] * pool[idx_k, l, :]
// One float4 per thread over 32*192 = 6144 float4s.
// ---------------------------------------------------------------------------
__global__ void combined_kernel(const float* __restrict__ pool,
                                const float* __restrict__ gates,
                                const int* __restrict__ idx,
                                float* __restrict__ combined) {
  const int C4 = EMBED / 4;                       // 192
  int i = blockIdx.x * blockDim.x + threadIdx.x;  // < 32*192
  if (i >= PLEN * C4) return;
  const int l = i / C4;
  const int d = i % C4;
  f4 acc = {0.f, 0.f, 0.f, 0.f};
#pragma unroll
  for (int k = 0; k < TOPK; ++k) {
    const float g = gates[k * PLEN + l];
    const f4 p = ((const f4*)pool)[(size_t)idx[k] * (PLEN * C4) + l * C4 + d];
    acc += g * p;
  }
  ((f4*)combined)[i] = acc;
}

// ---------------------------------------------------------------------------
// Kernel 5: assemble prompts[1000, 77, 768].
// grid (1000, 4): one class per blockIdx.x, quarter-row per blockIdx.y.
// combined/ctx are re-read 1000x -> stay hot in L2; output uses NT stores.
// ---------------------------------------------------------------------------
__global__ void assemble_kernel(const float* __restrict__ prefix,
                                const float* __restrict__ suffix,
                                const float* __restrict__ combined,
                                const float* __restrict__ ctx,
                                float* __restrict__ out) {
  const int C4   = EMBED / 4;       // 192
  const int ROW4 = 77 * C4;         // 14784 float4 per class
  const int CHNK = ROW4 / 4;        // 3696
  const int c = blockIdx.x;
  const f4* pre4 = (const f4*)prefix + (size_t)c * C4;
  const f4* suf4 = (const f4*)suffix + (size_t)c * SUFL * C4;
  const f4* com4 = (const f4*)combined;
  const f4* ctx4 = (const f4*)ctx;
  f4* o4 = (f4*)out + (size_t)c * ROW4;

  const int start = blockIdx.y * CHNK;
  const int end   = start + CHNK;
  for (int i = start + threadIdx.x; i < end; i += blockDim.x) {
    const int t = i / C4;
    const int d = i % C4;
    f4 v;
    if (t == 0)        v = pre4[d];
    else if (t <= 32)  v = com4[(t - 1) * C4 + d];
    else if (t <= 64)  v = ctx4[(t - 33) * C4 + d];
    else               v = suf4[(t - 65) * C4 + d];
    __builtin_nontemporal_store(v, &o4[i]);
  }
}

// ---------------------------------------------------------------------------
// Kernel 6: flat float4 copy (pass-through outputs).
// ---------------------------------------------------------------------------
__global__ void copy4_kernel(const float* __restrict__ src, float* __restrict__ dst, int n4) {
  int i = blockIdx.x * blockDim.x + threadIdx.x;
  if (i < n4) __builtin_nontemporal_store(((const f4*)src)[i], &((f4*)dst)[i]);
}

// ---------------------------------------------------------------------------
extern "C" void kernel_launch(void* const* d_in, const int* in_sizes, int n_in,
                              void* d_out, int out_size, void* d_ws, size_t ws_size,
                              hipStream_t stream) {
  const float* x_embed = (const float*)d_in[0];   // [16, 768]
  const float* pool    = (const float*)d_in[1];   // [32, 32, 768]
  const float* key     = (const float*)d_in[2];   // [32, 768]
  const float* aw      = (const float*)d_in[3];   // [768, 1]
  const float* ab      = (const float*)d_in[4];   // [1]
  const float* ctx     = (const float*)d_in[5];   // [32, 768]
  const float* prefix  = (const float*)d_in[6];   // [1000, 1, 768]
  const float* suffix  = (const float*)d_in[7];   // [1000, 12, 768]
  (void)d_in[8];                                  // train_flag: order-invariant

  // workspace layout (floats): q[768] | gates[160] | pad | combined[24576] | idx[5]
  float* W        = (float*)d_ws;
  float* ws_q     = W;
  float* ws_gates = W + 768;
  float* ws_comb  = W + 1024;
  int*   ws_idx   = (int*)(W + 1024 + PLEN * EMBED);

  float* out_prompts = (float*)d_out;                                   // 1000*77*768
  float* out_pool    = out_prompts + (size_t)NCLS * 77 * EMBED;         // +59,136,000
  float* out_key     = out_pool + POOL * PLEN * EMBED;                  // +786,432

  qnorm_kernel   <<<1, 256, 0, stream>>>(x_embed, ws_q);
  sim_topk_kernel<<<1,  64, 0, stream>>>(key, ws_q, ws_idx);
  gates_kernel   <<<1, 320, 0, stream>>>(pool, aw, ab, ws_idx, ws_gates);
  combined_kernel<<<24, 256, 0, stream>>>(pool, ws_gates, ws_idx, ws_comb);
  assemble_kernel<<<dim3(NCLS, 4), 256, 0, stream>>>(prefix, suffix, ws_comb, ctx, out_prompts);
  copy4_kernel   <<<(POOL * PLEN * EMBED / 4 + 255) / 256, 256, 0, stream>>>(pool, out_pool, POOL * PLEN * EMBED / 4);
  copy4_kernel   <<<(POOL * EMBED / 4 + 255) / 256, 256, 0, stream>>>(key, out_key, POOL * EMBED / 4);
}